// PairwiseConcat_24799141167443
// MI455X (gfx1250) — compile-verified
//
#include <hip/hip_runtime.h>

// Problem constants (from reference)
#define B_    2
#define L_    1024
#define HIN   768
#define HRED  128
#define HMLP  256
#define ROWS  (B_*L_)      // 2048 flattened (b,l) rows
#define NOUT  (2*HMLP)     // 512: U columns [0,256), V columns [256,512)

typedef __attribute__((ext_vector_type(2))) float v2f;
typedef __attribute__((ext_vector_type(8))) float v8f;

// ---------------------------------------------------------------------------
// Kernel 1: e[gi][r] = inputs[gi][:] . Wr[r][:] + br[r]
// [2048 x 768] x [768 -> 128], wave-per-16x16 tile, v_wmma_f32_16x16x4_f32.
// ---------------------------------------------------------------------------
__global__ __launch_bounds__(256) void gemm_e_kernel(
    const float* __restrict__ in, const float* __restrict__ Wr,
    const float* __restrict__ br, float* __restrict__ e) {
  const int wid  = threadIdx.x >> 5;
  const int lane = threadIdx.x & 31;
  const int t  = blockIdx.x * 8 + wid;   // 1024 wave-tiles total
  const int rt = t & 7;                  // r-tile (8 tiles of 16 over 128)
  const int it = t >> 3;                 // i-tile (128 tiles of 16 over 2048)

  const int l16 = lane & 15;
  const int kh  = (lane >> 4) << 1;      // K sub-offset {0,2}
  const int gi  = it * 16 + l16;         // A: M = lane%16
  const int r   = rt * 16 + l16;         // B/C: N = lane%16

  const float* arow = in + gi * HIN + kh;
  const float* brow = Wr + r  * HIN + kh;

  const float bias = br[r];              // N is uniform per lane across 8 C regs
  v8f acc;
  #pragma unroll
  for (int g = 0; g < 8; ++g) acc[g] = bias;

  #pragma unroll 4
  for (int k0 = 0; k0 < HIN; k0 += 4) {
    v2f a = *(const v2f*)(arow + k0);    // A[m][k0+kh .. +1], 8B aligned
    v2f b = *(const v2f*)(brow + k0);    // B[k][n] = Wr[n][k]
    acc = __builtin_amdgcn_wmma_f32_16x16x4_f32(
        false, a, false, b, (short)0, acc, false, false);
  }

  // C layout: lane holds C[M = g + 8*(lane/16)][N = lane%16]
  const int mrow0 = it * 16 + ((lane >> 4) << 3);
  #pragma unroll
  for (int g = 0; g < 8; ++g)
    e[(mrow0 + g) * HRED + r] = acc[g];
}

// ---------------------------------------------------------------------------
// Kernel 2: fused U/V GEMM, output transposed.
//   n' < 256 : OT[n'][gi] = e[gi][:] . W1[n'][0:128]   + b1[n']   (== U^T)
//   n' >=256 : OT[n'][gi] = e[gi][:] . W1[n'-256][128:256]        (== V^T)
// ---------------------------------------------------------------------------
__global__ __launch_bounds__(256) void gemm_uv_kernel(
    const float* __restrict__ e, const float* __restrict__ W1,
    const float* __restrict__ b1, float* __restrict__ OT) {
  const int wid  = threadIdx.x >> 5;
  const int lane = threadIdx.x & 31;
  const int t     = blockIdx.x * 8 + wid;  // 4096 wave-tiles total
  const int ntile = t & 31;                // 32 n'-tiles of 16 over 512
  const int itile = t >> 5;                // 128 i-tiles of 16 over 2048

  const int l16 = lane & 15;
  const int kh  = (lane >> 4) << 1;
  const int gi  = itile * 16 + l16;
  const int n   = ntile * 16 + l16;        // n' in [0,512); uniform side per wave

  const float* arow = e + gi * HRED + kh;
  const float* brow = (n < HMLP) ? (W1 + n * (2 * HRED) + kh)
                                 : (W1 + (n - HMLP) * (2 * HRED) + HRED + kh);
  const float bias = (n < HMLP) ? b1[n] : 0.0f;

  v8f acc;
  #pragma unroll
  for (int g = 0; g < 8; ++g) acc[g] = bias;

  #pragma unroll 4
  for (int k0 = 0; k0 < HRED; k0 += 4) {
    v2f a = *(const v2f*)(arow + k0);
    v2f b = *(const v2f*)(brow + k0);
    acc = __builtin_amdgcn_wmma_f32_16x16x4_f32(
        false, a, false, b, (short)0, acc, false, false);
  }

  // Transposed store: lane's 8 values share column n, consecutive rows gi.
  float* dst = OT + (size_t)n * ROWS + itile * 16 + ((lane >> 4) << 3);
  *(float4*)(dst)     = make_float4(acc[0], acc[1], acc[2], acc[3]);
  *(float4*)(dst + 4) = make_float4(acc[4], acc[5], acc[6], acc[7]);
}

// ---------------------------------------------------------------------------
// Kernel 3: epilogue. out[b,i,j] = b2 + sum_m w2[m]*relu(U[i,m]+V[j,m]),
// computed only for upper-triangular 64x64 tiles, mirrored into the lower
// triangle (matches triu(out) + triu(out,1)^T exactly).
// 4x4 micro-tile per thread: per m, 2x ds_load_b128 + 48 VALU ops; w2 comes
// in through scalar loads (uniform address), keeping the LDS pipe for u/v.
// ---------------------------------------------------------------------------
#define TS  64    // tile size in i and j
#define MC  64    // m-chunk staged in LDS (4 chunks of 64)
#define LDP 68    // padded LDS row stride (dwords): b128-aligned, conflict-free

__global__ __launch_bounds__(256) void epilogue_kernel(
    const float* __restrict__ OT, const float* __restrict__ W2,
    const float* __restrict__ b2, float* __restrict__ out) {
  __shared__ float Us[MC * LDP];
  __shared__ float Vs[MC * LDP];

  const int nt   = L_ / TS;            // 16
  const int ntri = nt * (nt + 1) / 2;  // 136
  const int tid  = threadIdx.x;

  int b = blockIdx.x / ntri;
  int p = blockIdx.x % ntri;
  int I = 0;
  while (p >= nt - I) { p -= nt - I; ++I; }
  const int J = I + p;                 // J >= I

  const int gi0 = b * L_ + I * TS;
  const int gj0 = b * L_ + J * TS;
  const float* Ug = OT + gi0;                       // U^T rows (m-major)
  const float* Vg = OT + (size_t)HMLP * ROWS + gj0; // V^T rows

  const int tx = tid & 15, ty = tid >> 4;           // 16x16 threads, 4x4 each
  float acc[4][4];
  #pragma unroll
  for (int a = 0; a < 4; ++a)
    #pragma unroll
    for (int c = 0; c < 4; ++c) acc[a][c] = 0.f;

  for (int mc = 0; mc < HMLP; mc += MC) {
    __syncthreads();
    // Stage MC x TS floats of U^T and V^T (coalesced b128 global loads).
    for (int q = tid; q < MC * (TS / 4); q += 256) {
      const int m  = q >> 4;           // TS/4 == 16
      const int i4 = (q & 15) << 2;
      *(float4*)&Us[m * LDP + i4] = *(const float4*)(Ug + (size_t)(mc + m) * ROWS + i4);
      *(float4*)&Vs[m * LDP + i4] = *(const float4*)(Vg + (size_t)(mc + m) * ROWS + i4);
    }
    __syncthreads();

    #pragma unroll 4
    for (int m = 0; m < MC; ++m) {
      const float  w = W2[mc + m];     // uniform address -> s_load (K$)
      const float4 u = *(const float4*)&Us[m * LDP + 4 * ty];
      const float4 v = *(const float4*)&Vs[m * LDP + 4 * tx];
      const float ua[4] = {u.x, u.y, u.z, u.w};
      const float va[4] = {v.x, v.y, v.z, v.w};
      #pragma unroll
      for (int a = 0; a < 4; ++a)
        #pragma unroll
        for (int c = 0; c < 4; ++c) {
          float h = ua[a] + va[c];
          h = h > 0.f ? h : 0.f;
          acc[a][c] += w * h;
        }
    }
  }

  const float bb = b2[0];
  #pragma unroll
  for (int a = 0; a < 4; ++a)
    #pragma unroll
    for (int c = 0; c < 4; ++c) acc[a][c] += bb;

  const int i0 = I * TS + 4 * ty;
  const int j0 = J * TS + 4 * tx;
  float* ob = out + (size_t)b * (L_ * L_);

  if (I < J) {
    // Entire tile strictly above the diagonal: vectorized direct + mirror.
    #pragma unroll
    for (int a = 0; a < 4; ++a)
      *(float4*)&ob[(size_t)(i0 + a) * L_ + j0] =
          make_float4(acc[a][0], acc[a][1], acc[a][2], acc[a][3]);
    #pragma unroll
    for (int c = 0; c < 4; ++c)
      *(float4*)&ob[(size_t)(j0 + c) * L_ + i0] =
          make_float4(acc[0][c], acc[1][c], acc[2][c], acc[3][c]);
  } else {
    // Diagonal tile: guarded scalar writes; (i>j) filled by mirror of (j,i).
    #pragma unroll
    for (int a = 0; a < 4; ++a)
      #pragma unroll
      for (int c = 0; c < 4; ++c) {
        const int i = i0 + a, j = j0 + c;
        if (i <= j) {
          ob[(size_t)i * L_ + j] = acc[a][c];
          if (i < j) ob[(size_t)j * L_ + i] = acc[a][c];
        }
      }
  }
}

// ---------------------------------------------------------------------------
extern "C" void kernel_launch(void* const* d_in, const int* in_sizes, int n_in,
                              void* d_out, int out_size, void* d_ws, size_t ws_size,
                              hipStream_t stream) {
  const float* in  = (const float*)d_in[0];  // [2,1024,768]
  const float* Wr  = (const float*)d_in[1];  // [128,768]
  const float* br  = (const float*)d_in[2];  // [128]
  const float* W1  = (const float*)d_in[3];  // [256,256]
  const float* b1  = (const float*)d_in[4];  // [256]
  const float* W2  = (const float*)d_in[5];  // [1,256]
  const float* b2  = (const float*)d_in[6];  // [1]
  float* out = (float*)d_out;                // [2,1024,1024]

  // Workspace layout: e (1 MB) | OT = [U^T ; V^T] (4 MB)
  float* e  = (float*)d_ws;
  float* OT = e + (size_t)ROWS * HRED;

  gemm_e_kernel <<<128,  256, 0, stream>>>(in, Wr, br, e);
  gemm_uv_kernel<<<512,  256, 0, stream>>>(e, W1, b1, OT);
  epilogue_kernel<<<B_ * (L_/TS) * (L_/TS + 1) / 2, 256, 0, stream>>>(OT, W2, b2, out);
}